// PFASModule_83897891160810
// MI455X (gfx1250) — compile-verified
//
#include <hip/hip_runtime.h>
#include <math.h>

typedef __attribute__((ext_vector_type(2))) float v2f;
typedef __attribute__((ext_vector_type(8))) float v8f;

#define NBATCH 8
#define PPTS   4096
#define NPTS   (NBATCH * PPTS)
#define KNN    32
#define CIN    64
#define COUT   64

// ---------------------------------------------------------------------------
// Kernel 0: point norms |p|^2 and zero the BN stat accumulators
// ---------------------------------------------------------------------------
__global__ __launch_bounds__(256) void init_kernel(const float* __restrict__ coord,
                                                   float* __restrict__ sqn,
                                                   float* __restrict__ stats)
{
    int i = blockIdx.x * blockDim.x + threadIdx.x;
    if (i < NPTS) {
        const float* c = coord + (size_t)i * 3;
        sqn[i] = c[0] * c[0] + c[1] * c[1] + c[2] * c[2];
    }
    if (i < 128) stats[i] = 0.0f;  // sum[64] + sumsq[64]
}

// ---------------------------------------------------------------------------
// Kernel 1: per-batch 32-NN + covariance eigen features (linearity, density)
// One wave per 16-query tile. The selection metric val = |c|^2 - 2*c.q is
// produced DIRECTLY by the WMMA by augmenting the K=4 slot:
//   A row m  = ( cx, cy, cz, |c|^2 )     (candidates)
//   B col n  = (-2qx,-2qy,-2qz, 1  )     (queries)
// D[m][n] lands with lane L owning query n=L%16, rows m = r + 8*(L/16):
// the lane pair (q, q+16) covers all 16 candidates of query q in registers.
// No LDS / barriers in the hot loop; a min-tree prefilter short-circuits the
// per-element insert path in the common (no-new-neighbor) case.
// ---------------------------------------------------------------------------
__global__ __launch_bounds__(256) void knn_geo_kernel(const float* __restrict__ coord,
                                                      const float* __restrict__ sqn,
                                                      float* __restrict__ linArr,
                                                      float* __restrict__ densArr)
{
    __shared__ float sMv[8][16][33];    // one-time merge buffers
    __shared__ int   sMi[8][16][33];

    const int lane = threadIdx.x & 31;
    const int wave = threadIdx.x >> 5;
    const int tile = blockIdx.x * 8 + wave;     // 0 .. 2047
    const int b    = tile >> 8;                 // batch
    const int q0   = (tile & 255) << 4;         // query tile base in batch
    const int base = b * PPTS;                  // batch point base (global)
    const int hi   = lane >> 4;
    const int ln   = lane & 15;

    // B operand (built once): VGPR0 = K(hi?2:0), VGPR1 = K(hi?3:1)
    const float* qc = coord + (size_t)(base + q0 + ln) * 3;
    v2f bop;
    if (hi) { bop.x = -2.0f * qc[2]; bop.y = 1.0f; }
    else    { bop.x = -2.0f * qc[0]; bop.y = -2.0f * qc[1]; }

    // A operand pointer streams (no branches in loop): lane<16 reads (x, y),
    // lane>=16 reads (z, |c|^2). Pointer-select once, stride-increment after.
    const float* pa0 = coord + (size_t)(base + ln) * 3 + (hi ? 2 : 0);
    const float* pa1 = hi ? (sqn + (size_t)(base + ln))
                          : (coord + (size_t)(base + ln) * 3 + 1);
    const int    st1 = hi ? 16 : 48;            // floats per 16-point tile

    // per-lane top-33 (smallest val) max-replace buffer
    float bval[33];
    int   bidx[33];
#pragma unroll
    for (int j = 0; j < 33; ++j) { bval[j] = 3.0e38f; bidx[j] = -1; }
    float curMax = 3.0e38f;
    int   curPos = 0;

    const int midx0 = 8 * hi;                   // this lane's first row in tile
    for (int c = 0; c < PPTS / 16; ++c) {
        v2f aop;
        aop.x = pa0[0];
        aop.y = pa1[0];
        pa0 += 48;
        pa1 += st1;

        v8f acc = {};
        acc = __builtin_amdgcn_wmma_f32_16x16x4_f32(false, aop, false, bop,
                                                    (short)0, acc, false, false);
        // acc[r] = |cand|^2 - 2*dot  for candidate m = r + 8*hi, query ln

        // min-tree prefilter: one compare/branch in the common case
        float m01 = fminf(acc[0], acc[1]);
        float m23 = fminf(acc[2], acc[3]);
        float m45 = fminf(acc[4], acc[5]);
        float m67 = fminf(acc[6], acc[7]);
        float m8  = fminf(fminf(m01, m23), fminf(m45, m67));
        if (m8 < curMax) {
#pragma unroll
            for (int r = 0; r < 8; ++r) {
                float val = acc[r];
                if (val < curMax) {
                    bval[curPos] = val;
                    bidx[curPos] = c * 16 + midx0 + r;
                    float mx = -3.0e38f; int mp = 0;
#pragma unroll 1
                    for (int j = 0; j < 33; ++j)
                        if (bval[j] > mx) { mx = bval[j]; mp = j; }
                    curMax = mx; curPos = mp;
                }
            }
        }
    }

    // upper lanes publish their buffers for the one-time merge
    if (hi) {
#pragma unroll 1
        for (int j = 0; j < 33; ++j) {
            sMv[wave][ln][j] = bval[j];
            sMi[wave][ln][j] = bidx[j];
        }
    }
    __syncthreads();

    if (!hi) {
        const int q   = ln;
        const int gq  = base + q0 + q;
        const float sqQ = sqn[gq];

        float sumd = 0.0f;
        float sx = 0, sy = 0, sz = 0;
        float sxx = 0, syy = 0, szz = 0, sxy = 0, sxz = 0, syz = 0;

        // extract the 33 smallest of 66; t==0 is self (global min of val)
        for (int t = 0; t < 33; ++t) {
            float bv = 3.0e38f; int bj = 0;
#pragma unroll 1
            for (int j = 0; j < 33; ++j)
                if (bval[j] < bv) { bv = bval[j]; bj = j; }
            int bj2 = -1;
#pragma unroll 1
            for (int j = 0; j < 33; ++j)
                if (sMv[wave][q][j] < bv) { bv = sMv[wave][q][j]; bj2 = j; }
            int nid;
            if (bj2 >= 0) { nid = sMi[wave][q][bj2]; sMv[wave][q][bj2] = 3.0e38f; }
            else          { nid = bidx[bj];          bval[bj]          = 3.0e38f; }
            if (t == 0) continue;  // drop self

            float d = sqrtf(fmaxf(bv + sqQ, 1e-12f));
            sumd += d;
            const float* pc = coord + (size_t)(base + nid) * 3;
            float px = pc[0], py = pc[1], pz = pc[2];
            sx += px; sy += py; sz += pz;
            sxx += px * px; syy += py * py; szz += pz * pz;
            sxy += px * py; sxz += px * pz; syz += py * pz;
        }

        const float inv32 = 1.0f / 32.0f, i31 = 1.0f / 31.0f;
        float mx = sx * inv32, my = sy * inv32, mz = sz * inv32;
        float cxx = (sxx - 32.0f * mx * mx) * i31;
        float cyy = (syy - 32.0f * my * my) * i31;
        float czz = (szz - 32.0f * mz * mz) * i31;
        float cxy = (sxy - 32.0f * mx * my) * i31;
        float cxz = (sxz - 32.0f * mx * mz) * i31;
        float cyz = (syz - 32.0f * my * mz) * i31;

        // closed-form largest eigenvalue of symmetric 3x3
        float tr = cxx + cyy + czz;
        float qd = tr * (1.0f / 3.0f);
        float p1 = cxy * cxy + cxz * cxz + cyz * cyz;
        float aa = cxx - qd, bb = cyy - qd, cc2 = czz - qd;
        float p2 = aa * aa + bb * bb + cc2 * cc2 + 2.0f * p1;
        float pp = sqrtf(fmaxf(p2 * (1.0f / 6.0f), 0.0f));
        float e0 = qd;
        if (pp > 1e-20f) {
            float ip = 1.0f / pp;
            float b00 = aa * ip, b11 = bb * ip, b22 = cc2 * ip;
            float b01 = cxy * ip, b02 = cxz * ip, b12 = cyz * ip;
            float detB = b00 * (b11 * b22 - b12 * b12)
                       - b01 * (b01 * b22 - b12 * b02)
                       + b02 * (b01 * b12 - b11 * b02);
            float r = fminf(fmaxf(detB * 0.5f, -1.0f), 1.0f);
            float phi = acosf(r) * (1.0f / 3.0f);
            e0 = qd + 2.0f * pp * cosf(phi);
        }
        float trs = fmaxf(tr, 1e-30f);
        linArr[gq]  = (2.0f * e0 - tr) / trs;      // ev0 - (ev1 + ev2)
        densArr[gq] = 1.0f / (sumd * inv32 + 1e-6f);
    }
}

// ---------------------------------------------------------------------------
// Kernel 2: h = feat @ W1^T + b1 via WMMA f32 16x16x4, with BN stat partials
// One wave per 16-row tile of h; 16 k-steps x 4 n-tiles = 64 WMMAs / wave.
// ---------------------------------------------------------------------------
__global__ __launch_bounds__(256) void gemm_bn_kernel(const float* __restrict__ feat,
                                                      const float* __restrict__ W1,
                                                      const float* __restrict__ b1,
                                                      float* __restrict__ h,
                                                      float* __restrict__ stats)
{
    const int lane = threadIdx.x & 31;
    const int wave = threadIdx.x >> 5;
    const int tile = blockIdx.x * 8 + wave;   // 0 .. 2047
    const int m0   = tile * 16;
    const int hi   = lane >> 4;
    const int ln   = lane & 15;

    v8f acc0 = {}, acc1 = {}, acc2 = {}, acc3 = {};
    const float* arow = feat + (size_t)(m0 + ln) * CIN + 2 * hi;   // A: feat rows
    const float* brow = W1 + (size_t)ln * CIN + 2 * hi;            // B: W1 rows

#pragma unroll 4
    for (int ks = 0; ks < 16; ++ks) {
        v2f a  = *(const v2f*)(arow + ks * 4);
        v2f b0 = *(const v2f*)(brow + 0 * 16 * CIN + ks * 4);
        v2f b1v= *(const v2f*)(brow + 1 * 16 * CIN + ks * 4);
        v2f b2v= *(const v2f*)(brow + 2 * 16 * CIN + ks * 4);
        v2f b3v= *(const v2f*)(brow + 3 * 16 * CIN + ks * 4);
        acc0 = __builtin_amdgcn_wmma_f32_16x16x4_f32(false, a, false, b0,  (short)0, acc0, false, false);
        acc1 = __builtin_amdgcn_wmma_f32_16x16x4_f32(false, a, false, b1v, (short)0, acc1, false, false);
        acc2 = __builtin_amdgcn_wmma_f32_16x16x4_f32(false, a, false, b2v, (short)0, acc2, false, false);
        acc3 = __builtin_amdgcn_wmma_f32_16x16x4_f32(false, a, false, b3v, (short)0, acc3, false, false);
    }

    // epilogue: bias, store h, per-channel BN partial sums (lane owns channel)
    auto epi = [&](const v8f& acc, int nt) {
        int ch = nt * 16 + ln;
        float bias = b1[ch];
        float s = 0.0f, s2 = 0.0f;
#pragma unroll
        for (int r = 0; r < 8; ++r) {
            float v = acc[r] + bias;
            h[(size_t)(m0 + r + 8 * hi) * COUT + ch] = v;
            s += v; s2 += v * v;
        }
        atomicAdd(&stats[ch], s);
        atomicAdd(&stats[64 + ch], s2);
    };
    epi(acc0, 0); epi(acc1, 1); epi(acc2, 2); epi(acc3, 3);
}

// ---------------------------------------------------------------------------
// Kernel 3: batchnorm + ReLU + 64->3 head + softmax + GRID combine
// ---------------------------------------------------------------------------
__global__ __launch_bounds__(256) void head_kernel(const float* __restrict__ h,
                                                   const float* __restrict__ stats,
                                                   const float* __restrict__ gamma,
                                                   const float* __restrict__ beta,
                                                   const float* __restrict__ W2,
                                                   const float* __restrict__ b2,
                                                   const float* __restrict__ linArr,
                                                   const float* __restrict__ densArr,
                                                   float* __restrict__ out)
{
    __shared__ float sMu[64], sRs[64], sG[64], sB[64], sW2[3][64];
    int t = threadIdx.x;
    if (t < 64) {
        float mu  = stats[t] * (1.0f / (float)NPTS);
        float var = stats[64 + t] * (1.0f / (float)NPTS) - mu * mu;
        sMu[t] = mu;
        sRs[t] = rsqrtf(var + 1e-5f);
        sG[t]  = gamma[t];
        sB[t]  = beta[t];
        sW2[0][t] = W2[t]; sW2[1][t] = W2[64 + t]; sW2[2][t] = W2[128 + t];
    }
    __syncthreads();

    int i = blockIdx.x * blockDim.x + t;
    if (i >= NPTS) return;

    float l0 = b2[0], l1 = b2[1], l2 = b2[2];
    const float* hrow = h + (size_t)i * COUT;
#pragma unroll 4
    for (int c = 0; c < 64; ++c) {
        float v = (hrow[c] - sMu[c]) * sRs[c] * sG[c] + sB[c];
        v = fmaxf(v, 0.0f);
        l0 = fmaf(v, sW2[0][c], l0);
        l1 = fmaf(v, sW2[1][c], l1);
        l2 = fmaf(v, sW2[2][c], l2);
    }
    float mxl = fmaxf(l0, fmaxf(l1, l2));
    float e0 = expf(l0 - mxl), e1 = expf(l1 - mxl), e2 = expf(l2 - mxl);
    float is = 1.0f / (e0 + e1 + e2);
    float p0 = e0 * is, p1 = e1 * is, p2 = e2 * is;

    float lin = linArr[i], dens = densArr[i];
    float tower = (dens * 2.0f + p0) * (1.0f / 3.0f);
    float bg    = (fmaxf(1.0f - lin, 1.0f - dens) + p1) * (1.0f / 3.0f);
    float line  = (lin * 2.0f + p2) * (1.0f / 3.0f);

    // GRID rows: tower->0.05, background->0.5, line_grid = [0.1, 0.1, 0.5]
    float o01 = tower * 0.05f + bg * 0.5f + line * 0.1f + 1e-6f;
    float o2  = tower * 0.05f + bg * 0.5f + line * 0.5f + 1e-6f;
    out[(size_t)i * 3 + 0] = o01;
    out[(size_t)i * 3 + 1] = o01;
    out[(size_t)i * 3 + 2] = o2;
}

// ---------------------------------------------------------------------------
extern "C" void kernel_launch(void* const* d_in, const int* in_sizes, int n_in,
                              void* d_out, int out_size, void* d_ws, size_t ws_size,
                              hipStream_t stream)
{
    const float* feat  = (const float*)d_in[0];
    const float* coord = (const float*)d_in[1];
    // d_in[2] = batch (int64) — implied by layout, unused
    const float* W1    = (const float*)d_in[3];
    const float* b1    = (const float*)d_in[4];
    const float* gamma = (const float*)d_in[5];
    const float* beta  = (const float*)d_in[6];
    const float* W2    = (const float*)d_in[7];
    const float* b2    = (const float*)d_in[8];
    float* out = (float*)d_out;

    float* ws    = (float*)d_ws;
    float* sqn   = ws;                          // NPTS
    float* linA  = ws + NPTS;                   // NPTS
    float* densA = ws + 2 * (size_t)NPTS;       // NPTS
    float* stats = ws + 3 * (size_t)NPTS;       // 128 (sum, sumsq)
    float* h     = ws + 3 * (size_t)NPTS + 128; // NPTS * 64

    init_kernel<<<(NPTS + 255) / 256, 256, 0, stream>>>(coord, sqn, stats);
    knn_geo_kernel<<<256, 256, 0, stream>>>(coord, sqn, linA, densA);
    gemm_bn_kernel<<<256, 256, 0, stream>>>(feat, W1, b1, h, stats);
    head_kernel<<<(NPTS + 255) / 256, 256, 0, stream>>>(h, stats, gamma, beta, W2,
                                                        b2, linA, densA, out);
}